// DC_and_BD_loss_29712583753731
// MI455X (gfx1250) — compile-verified
//
#include <hip/hip_runtime.h>

typedef __attribute__((ext_vector_type(2))) float v2f;
typedef __attribute__((ext_vector_type(8))) float v8f;

#define TPB 256
#define NBLK 1024

// D = A(16x4) x B(4x16) + C. f32 layouts: A[m][k] lives at lane (m + 16*(k>=2)),
// VGPR (k&1); B[k][n] mirrors it (half-lane row striping, same as C/D).
// => lane L's (a.x*b.x + a.y*b.y) lands on diagonal D[n][n], n = L%16.
// trace(D) = sum over all 32 lanes of the per-lane 2-element dot products.
__device__ __forceinline__ v8f wmma_acc(v8f c, float a0, float a1, float b0, float b1) {
    v2f a, b;
    a.x = a0; a.y = a1;
    b.x = b0; b.y = b1;
    return __builtin_amdgcn_wmma_f32_16x16x4_f32(false, a, false, b, (short)0, c,
                                                 false, false);
}

struct Dice9 {
    float T1, T2, T3, Sp1, Sp2, Sp3, Sy1, Sy2, Sy3;
};

// Process 4 consecutive voxels starting at vb. GUARD=false => no branch at all.
template <bool GUARD>
__device__ __forceinline__ void process4(const float* __restrict__ net,
                                         const int* __restrict__ tgt,
                                         const float* __restrict__ phi,
                                         int vb, int N, bool valid, Dice9& s,
                                         float p1[4], float p2[4], float p3[4],
                                         float f1[4], float f2[4], float f3[4]) {
    if (GUARD) {
#pragma unroll
        for (int j = 0; j < 4; ++j) {
            p1[j] = p2[j] = p3[j] = 0.f;
            f1[j] = f2[j] = f3[j] = 0.f;
        }
        if (!valid) return;
    }
    const int b = (vb >= N) ? 1 : 0;        // B == 2 (reference shape)
    const int base = vb + b * 3 * N;        // flat index of (b, c=0, voxel vb)
    const float4 x0 = *(const float4*)(net + base);
    const float4 x1 = *(const float4*)(net + base + N);
    const float4 x2 = *(const float4*)(net + base + 2 * N);
    const float4 x3 = *(const float4*)(net + base + 3 * N);
    const float4 g1 = *(const float4*)(phi + base + N);
    const float4 g2 = *(const float4*)(phi + base + 2 * N);
    const float4 g3 = *(const float4*)(phi + base + 3 * N);
    const int4 t4 = *(const int4*)(tgt + vb);

    auto vox = [&](float a0, float a1, float a2, float a3,
                   float h1, float h2, float h3, int t, int j) {
        float m = fmaxf(fmaxf(a0, a1), fmaxf(a2, a3));
        float e0 = __expf(a0 - m), e1 = __expf(a1 - m);
        float e2 = __expf(a2 - m), e3 = __expf(a3 - m);
        // hardware v_rcp_f32 (1 ulp) instead of IEEE div expansion
        float inv = __builtin_amdgcn_rcpf(e0 + e1 + e2 + e3);
        float q1 = e1 * inv, q2 = e2 * inv, q3 = e3 * inv;
        s.Sp1 += q1; s.Sp2 += q2; s.Sp3 += q3;
        // branchless one-hot accumulation (cndmask, no exec masking)
        float w1 = (t == 1) ? 1.f : 0.f;
        float w2 = (t == 2) ? 1.f : 0.f;
        float w3 = (t == 3) ? 1.f : 0.f;
        s.Sy1 += w1; s.Sy2 += w2; s.Sy3 += w3;
        s.T1 += w1 * q1; s.T2 += w2 * q2; s.T3 += w3 * q3;
        p1[j] = q1; p2[j] = q2; p3[j] = q3;
        f1[j] = h1; f2[j] = h2; f3[j] = h3;
    };
    vox(x0.x, x1.x, x2.x, x3.x, g1.x, g2.x, g3.x, t4.x, 0);
    vox(x0.y, x1.y, x2.y, x3.y, g1.y, g2.y, g3.y, t4.y, 1);
    vox(x0.z, x1.z, x2.z, x3.z, g1.z, g2.z, g3.z, t4.z, 2);
    vox(x0.w, x1.w, x2.w, x3.w, g1.w, g2.w, g3.w, t4.w, 3);
}

__global__ void loss_main_kernel(const float* __restrict__ net,
                                 const int* __restrict__ tgt,
                                 const float* __restrict__ phi,
                                 float* __restrict__ partial,
                                 int nvec, int N) {
    const int tid = blockIdx.x * blockDim.x + threadIdx.x;
    const int nthreads = gridDim.x * blockDim.x;
    const int full = nvec / nthreads;       // uniform, unguarded iterations

    Dice9 s = {0.f, 0.f, 0.f, 0.f, 0.f, 0.f, 0.f, 0.f, 0.f};
    v8f acc0 = {0.f, 0.f, 0.f, 0.f, 0.f, 0.f, 0.f, 0.f};
    v8f acc1 = acc0;
    float p1[4], p2[4], p3[4], f1[4], f2[4], f3[4];

    for (int it = 0; it < full; ++it) {
        const int vb = (tid + it * nthreads) * 4;
        process4<false>(net, tgt, phi, vb, N, true, s, p1, p2, p3, f1, f2, f3);
        acc0 = wmma_acc(acc0, p1[0], p1[1], f1[0], f1[1]);
        acc0 = wmma_acc(acc0, p1[2], p1[3], f1[2], f1[3]);
        acc0 = wmma_acc(acc0, p2[0], p2[1], f2[0], f2[1]);
        acc0 = wmma_acc(acc0, p2[2], p2[3], f2[2], f2[3]);
        acc1 = wmma_acc(acc1, p3[0], p3[1], f3[0], f3[1]);
        acc1 = wmma_acc(acc1, p3[2], p3[3], f3[2], f3[3]);
    }
    const int restbase = full * nthreads;
    if (restbase < nvec) {                  // scalar-uniform branch
        const int v4 = restbase + tid;
        process4<true>(net, tgt, phi, v4 * 4, N, v4 < nvec, s,
                       p1, p2, p3, f1, f2, f3);
        // EXEC is full again here (guard reconverged inside process4)
        acc0 = wmma_acc(acc0, p1[0], p1[1], f1[0], f1[1]);
        acc0 = wmma_acc(acc0, p1[2], p1[3], f1[2], f1[3]);
        acc0 = wmma_acc(acc0, p2[0], p2[1], f2[0], f2[1]);
        acc0 = wmma_acc(acc0, p2[2], p2[3], f2[2], f2[3]);
        acc1 = wmma_acc(acc1, p3[0], p3[1], f3[0], f3[1]);
        acc1 = wmma_acc(acc1, p3[2], p3[3], f3[2], f3[3]);
    }

    // extract this lane's diagonal (trace) contribution
    v8f accs = acc0 + acc1;
    union { v8f v; float f[8]; } u;
    u.v = accs;
    const int lane = threadIdx.x & 31;
    float dval = 0.f;
    if (lane < 8)        dval = u.f[lane];
    else if (lane >= 24) dval = u.f[lane - 24];

    float vals[10] = {s.T1, s.T2, s.T3, s.Sp1, s.Sp2, s.Sp3,
                      s.Sy1, s.Sy2, s.Sy3, dval};
#pragma unroll
    for (int off = 16; off > 0; off >>= 1) {
#pragma unroll
        for (int k = 0; k < 10; ++k)
            vals[k] += __shfl_xor(vals[k], off, 32);
    }

    __shared__ float red[TPB / 32][10];
    const int w = threadIdx.x >> 5;
    if (lane == 0) {
#pragma unroll
        for (int k = 0; k < 10; ++k) red[w][k] = vals[k];
    }
    __syncthreads();
    if (threadIdx.x < 10) {
        float acc = 0.f;
#pragma unroll
        for (int ww = 0; ww < TPB / 32; ++ww) acc += red[ww][threadIdx.x];
        partial[blockIdx.x * 10 + threadIdx.x] = acc;
    }
}

__global__ void loss_final_kernel(const float* __restrict__ partial,
                                  float* __restrict__ out,
                                  int nb, float bd_count) {
    float loc[10] = {0.f, 0.f, 0.f, 0.f, 0.f, 0.f, 0.f, 0.f, 0.f, 0.f};
    for (int r = threadIdx.x; r < nb; r += blockDim.x) {
#pragma unroll
        for (int k = 0; k < 10; ++k) loc[k] += partial[r * 10 + k];
    }
    const int lane = threadIdx.x & 31;
    const int w = threadIdx.x >> 5;
#pragma unroll
    for (int off = 16; off > 0; off >>= 1) {
#pragma unroll
        for (int k = 0; k < 10; ++k) loc[k] += __shfl_xor(loc[k], off, 32);
    }
    __shared__ float red[8][10];
    if (lane == 0) {
#pragma unroll
        for (int k = 0; k < 10; ++k) red[w][k] = loc[k];
    }
    __syncthreads();
    if (threadIdx.x == 0) {
        float v[10];
#pragma unroll
        for (int k = 0; k < 10; ++k) {
            float acc = 0.f;
#pragma unroll
            for (int ww = 0; ww < 8; ++ww) acc += red[ww][k];
            v[k] = acc;
        }
        const float S = 1e-5f;
        // dc = (2T+S) / (2T + (Sp-T) + (Sy-T) + S) = (2T+S)/(Sp+Sy+S)
        float dc1 = (2.f * v[0] + S) / (v[3] + v[6] + S);
        float dc2 = (2.f * v[1] + S) / (v[4] + v[7] + S);
        float dc3 = (2.f * v[2] + S) / (v[5] + v[8] + S);
        float dc_loss = 1.f - (dc1 + dc2 + dc3) * (1.f / 3.f);
        float bd_loss = v[9] / bd_count;
        out[0] = 0.01f * dc_loss + 0.99f * bd_loss;
    }
}

extern "C" void kernel_launch(void* const* d_in, const int* in_sizes, int n_in,
                              void* d_out, int out_size, void* d_ws, size_t ws_size,
                              hipStream_t stream) {
    const float* net = (const float*)d_in[0];   // [B,C,X,Y,Z] f32
    const int* tgt = (const int*)d_in[1];       // [B,1,X,Y,Z] i32
    const float* phi = (const float*)d_in[2];   // [B,C,X,Y,Z] f32
    float* out = (float*)d_out;
    float* partial = (float*)d_ws;

    const int V = in_sizes[1];      // B*N voxels (B=2 per reference)
    const int B = 2;
    const int N = V / B;            // X*Y*Z
    const int nvec = V / 4;         // float4 groups

    int nb = NBLK;                  // deterministic clamp to available scratch
    while ((size_t)nb * 10 * sizeof(float) > ws_size && nb > 64) nb >>= 1;

    loss_main_kernel<<<nb, TPB, 0, stream>>>(net, tgt, phi, partial, nvec, N);
    loss_final_kernel<<<1, 256, 0, stream>>>(partial, out, nb, 3.0f * (float)V);
}